// SlicedWassersteinDist_55061480734989
// MI455X (gfx1250) — compile-verified
//
#include <hip/hip_runtime.h>
#include <math.h>

typedef __attribute__((ext_vector_type(2))) float v2f;
typedef __attribute__((ext_vector_type(4))) float v4f;
typedef __attribute__((ext_vector_type(8))) float v8f;

constexpr int B  = 128;
constexpr int N  = 2048;
constexpr int D  = 3;
constexpr int L  = 100;
constexpr int LP = 112;          // L padded to 7 tiles of 16 for WMMA
constexpr int NT = N / 16;       // 128 n-tiles of 16 points

// ---------------------------------------------------------------------------
// Kernel 1: projection via V_WMMA_F32_16X16X4_F32.
// One wave (32 threads) per (n-tile, chunk-batch). Computes a 16x16 tile of
// Pp and Qp for 7 L-tiles, writing TRANSPOSED [cb][l][n] so the sort kernel
// reads contiguous columns.
//
// A-matrix 16x4 f32 layout (ISA 7.12.2): lanes 0-15 hold K=0,1 in VGPR0,1 for
// row M=lane; lanes 16-31 hold K=2,3 for row M=lane-16. K=3 (D=3 padded to 4)
// is zero. B-matrix mirrors this with N=lane%16 columns.
// D-matrix: VGPR r, lanes 0-15 -> M=r, N=lane; lanes 16-31 -> M=8+r, N=lane-16
// => each lane owns 8 consecutive n for one l column: two float4 stores.
// ---------------------------------------------------------------------------
__global__ __launch_bounds__(32)
void swd_project_wmma(const float* __restrict__ P, const float* __restrict__ Q,
                      const float* __restrict__ proj,
                      float* __restrict__ wsP, float* __restrict__ wsQ,
                      int bBase)
{
    const int lane = threadIdx.x;
    const int m    = lane & 15;            // row (A) / col (B) within tile
    const int k0   = (lane >> 4) << 1;     // 0 for lanes 0-15, 2 for 16-31
    const int n0   = blockIdx.x * 16;      // point-tile base
    const int cb   = blockIdx.y;           // chunk-local batch
    const int b    = bBase + cb;           // global batch

    // A tiles (reused across all 7 L-tiles)
    const float* prow = P + ((size_t)b * N + (n0 + m)) * D;
    const float* qrow = Q + ((size_t)b * N + (n0 + m)) * D;
    v2f aP, aQ;
    aP.x = prow[k0];
    aP.y = (k0 + 1 < D) ? prow[k0 + 1] : 0.0f;   // K=3 pad -> 0
    aQ.x = qrow[k0];
    aQ.y = (k0 + 1 < D) ? qrow[k0 + 1] : 0.0f;

    const float* pb = proj + (size_t)b * D * L;  // [d][l], row-major

    #pragma unroll
    for (int lt = 0; lt < LP / 16; ++lt) {
        const int l = lt * 16 + m;
        v2f bv;
        bv.x = (l < L) ? pb[(size_t)k0 * L + l] : 0.0f;
        bv.y = (l < L && (k0 + 1) < D) ? pb[(size_t)(k0 + 1) * L + l] : 0.0f;

        v8f cz = {};
        // (neg_a, A, neg_b, B, c_mod, C, reuse_a, reuse_b)
        v8f cP = __builtin_amdgcn_wmma_f32_16x16x4_f32(
            false, aP, false, bv, (short)0, cz, false, false);
        v8f cQ = __builtin_amdgcn_wmma_f32_16x16x4_f32(
            false, aQ, false, bv, (short)0, cz, false, false);

        const int nb = n0 + ((lane >> 4) << 3);   // this lane's first row
        float* dP = wsP + ((size_t)cb * LP + l) * N + nb;
        float* dQ = wsQ + ((size_t)cb * LP + l) * N + nb;
        v4f p0 = { cP[0], cP[1], cP[2], cP[3] };
        v4f p1 = { cP[4], cP[5], cP[6], cP[7] };
        v4f q0 = { cQ[0], cQ[1], cQ[2], cQ[3] };
        v4f q1 = { cQ[4], cQ[5], cQ[6], cQ[7] };
        *(v4f*)(dP)     = p0;
        *(v4f*)(dP + 4) = p1;
        *(v4f*)(dQ)     = q0;
        *(v4f*)(dQ + 4) = q1;
    }
}

// ---------------------------------------------------------------------------
// Kernel 2: per-(b,l) column sort (bitonic, in LDS) of Pp and Qp, then
// sum of squared sorted differences -> partial[b*L + l]. 256 threads,
// 8 elements/thread, both arrays share the same barrier schedule.
// ---------------------------------------------------------------------------
__global__ __launch_bounds__(256)
void swd_sort_diff(const float* __restrict__ wsP, const float* __restrict__ wsQ,
                   float* __restrict__ partial, int bBase)
{
    __shared__ float sP[N];
    __shared__ float sQ[N];
    __shared__ float red[256];

    const int tid = threadIdx.x;
    const int l   = blockIdx.x;        // 0..L-1
    const int cb  = blockIdx.y;        // chunk-local batch
    const size_t base = ((size_t)cb * LP + l) * N;

    for (int i = tid; i < N; i += 256) {
        sP[i] = wsP[base + i];
        sQ[i] = wsQ[base + i];
    }
    __syncthreads();

    // bitonic sort, ascending
    for (int k = 2; k <= N; k <<= 1) {
        for (int j = k >> 1; j > 0; j >>= 1) {
            for (int i = tid; i < N; i += 256) {
                const int ixj = i ^ j;
                if (ixj > i) {
                    const bool up = ((i & k) == 0);
                    float a = sP[i], b2 = sP[ixj];
                    if ((a > b2) == up) { sP[i] = b2; sP[ixj] = a; }
                    float c = sQ[i], d2 = sQ[ixj];
                    if ((c > d2) == up) { sQ[i] = d2; sQ[ixj] = c; }
                }
            }
            __syncthreads();
        }
    }

    float acc = 0.0f;
    for (int i = tid; i < N; i += 256) {
        const float d = sP[i] - sQ[i];
        acc += d * d;
    }
    red[tid] = acc;
    __syncthreads();
    #pragma unroll
    for (int s = 128; s > 0; s >>= 1) {
        if (tid < s) red[tid] += red[tid + s];
        __syncthreads();
    }
    if (tid == 0) partial[(size_t)(bBase + cb) * L + l] = red[0];
}

// ---------------------------------------------------------------------------
// Kernel 3: per-batch mean over (N,L), sqrt, then mean over B. Deterministic
// serial sums (no float atomics anywhere).
// ---------------------------------------------------------------------------
__global__ __launch_bounds__(128)
void swd_finalize(const float* __restrict__ partial, float* __restrict__ out)
{
    __shared__ float red[128];
    const int b = threadIdx.x;
    float s = 0.0f;
    for (int l = 0; l < L; ++l) s += partial[(size_t)b * L + l];
    red[b] = sqrtf(s * (1.0f / (float)(N * L)));
    __syncthreads();
    #pragma unroll
    for (int t = 64; t > 0; t >>= 1) {
        if (b < t) red[b] += red[b + t];
        __syncthreads();
    }
    if (b == 0) out[0] = red[0] * (1.0f / (float)B);
}

// ---------------------------------------------------------------------------
// Launch: chunk batches so the [chunk][LP][N] x2 workspace fits ws_size (and,
// for chunk<=16, stays resident in the 192MB L2 between project and sort).
// ---------------------------------------------------------------------------
extern "C" void kernel_launch(void* const* d_in, const int* in_sizes, int n_in,
                              void* d_out, int out_size, void* d_ws, size_t ws_size,
                              hipStream_t stream)
{
    const float* P    = (const float*)d_in[0];
    const float* Q    = (const float*)d_in[1];
    const float* proj = (const float*)d_in[2];

    float* partial = (float*)d_ws;                       // B*L floats
    const size_t partialBytes = (size_t)B * L * sizeof(float);
    char* dataBase = (char*)d_ws + partialBytes;

    const size_t perBatchFloats = (size_t)LP * N;        // one of P/Q
    const size_t bytesPerBatch  = perBatchFloats * sizeof(float) * 2;

    size_t avail = (ws_size > partialBytes) ? (ws_size - partialBytes) : 0;
    int chunk = (int)(avail / bytesPerBatch);
    if (chunk < 1)  chunk = 1;    // assume ws covers at least one batch (1.84MB)
    if (chunk > B)  chunk = B;

    float* wsP = (float*)dataBase;
    float* wsQ = wsP + (size_t)chunk * perBatchFloats;

    for (int b0 = 0; b0 < B; b0 += chunk) {
        const int nb = (B - b0 < chunk) ? (B - b0) : chunk;
        swd_project_wmma<<<dim3(NT, nb), 32, 0, stream>>>(P, Q, proj, wsP, wsQ, b0);
        swd_sort_diff<<<dim3(L, nb), 256, 0, stream>>>(wsP, wsQ, partial, b0);
    }
    swd_finalize<<<1, 128, 0, stream>>>(partial, (float*)d_out);
}